// FreeEnergyBound_65051574665800
// MI455X (gfx1250) — compile-verified
//
#include <hip/hip_runtime.h>

// FreeEnergyBound on MI455X (gfx1250, wave32).
// Pure bandwidth-bound global reduction: 512MB NT-streamed reads, two-stage
// deterministic reduce. Wave-level sums use V_WMMA_F32_16X16X4_F32.

typedef float v2f __attribute__((ext_vector_type(2)));
typedef float v4f __attribute__((ext_vector_type(4)));
typedef float v8f __attribute__((ext_vector_type(8)));

#define NBLK 2048
#define TPB  256
#define BATCH 65536u   // B
#define ZDIM  512u     // Z

__device__ __forceinline__ v4f ntload4(const float* p) {
    return __builtin_nontemporal_load((const v4f*)p);
}

// Wave-wide (32-lane) sum of two independent accumulators via WMMA.
// A-matrix (16x4 f32, 2 VGPRs/lane): lane L holds A[L%16][{0or2}] = x,
// A[L%16][{1or3}] = y. B = column selector: {1,0} picks K=0,2 (the x's),
// {0,1} picks K=1,3 (the y's). D row m = x[m]+x[m+16] (resp. y).
// Summing the 8 D VGPRs gives the half-wave total; xor-16 shuffle finishes.
// EXEC must be all ones at the call site (callers are convergent).
__device__ __forceinline__ void wave_sum2(float x, float y, float& sx, float& sy) {
    v2f a;   a.x   = x;    a.y   = y;
    v2f b10; b10.x = 1.0f; b10.y = 0.0f;
    v2f b01; b01.x = 0.0f; b01.y = 1.0f;
    v8f zero = {0.f, 0.f, 0.f, 0.f, 0.f, 0.f, 0.f, 0.f};
    v8f dx = __builtin_amdgcn_wmma_f32_16x16x4_f32(false, a, false, b10,
                                                   (short)0, zero, false, false);
    v8f dy = __builtin_amdgcn_wmma_f32_16x16x4_f32(false, a, false, b01,
                                                   (short)0, zero, false, false);
    float px = ((dx[0] + dx[1]) + (dx[2] + dx[3])) + ((dx[4] + dx[5]) + (dx[6] + dx[7]));
    float py = ((dy[0] + dy[1]) + (dy[2] + dy[3])) + ((dy[4] + dy[5]) + (dy[6] + dy[7]));
    sx = px + __shfl_xor(px, 16, 32);
    sy = py + __shfl_xor(py, 16, 32);
}

// Block-wide sum of 4 accumulators; thread 0 writes the 4 results to out4.
__device__ __forceinline__ void block_sum4(float a0, float a1, float a2, float a3,
                                           float* __restrict__ out4) {
    __shared__ float sm[4][TPB / 32];
    float s0, s1, s2, s3;
    wave_sum2(a0, a1, s0, s1);
    wave_sum2(a2, a3, s2, s3);
    const int wave = threadIdx.x >> 5;
    const int lane = threadIdx.x & 31;
    if (lane == 0) {
        sm[0][wave] = s0; sm[1][wave] = s1; sm[2][wave] = s2; sm[3][wave] = s3;
    }
    __syncthreads();
    if (threadIdx.x == 0) {
        float t0 = 0.f, t1 = 0.f, t2 = 0.f, t3 = 0.f;
        #pragma unroll
        for (int w = 0; w < TPB / 32; ++w) {
            t0 += sm[0][w]; t1 += sm[1][w]; t2 += sm[2][w]; t3 += sm[3][w];
        }
        out4[0] = t0; out4[1] = t1; out4[2] = t2; out4[3] = t3;
    }
}

// Stage 1: stream T (B,2,Z) and nat_params (B,Z,2), accumulate per-block
// partials {trace, sum_log2|eta0|, sum eta1^2/(4 eta0), sum log_jacobians}.
__global__ void __launch_bounds__(TPB)
fe_stage1(const float* __restrict__ T, const float* __restrict__ NP,
          const float* __restrict__ LJ, float* __restrict__ part) {
    const unsigned NU     = BATCH * (ZDIM / 4u);          // 8,388,608 quad-z units
    const unsigned stride = gridDim.x * blockDim.x;        // 524,288 -> 16 iters/thread
    float tr = 0.f, lg = 0.f, rt = 0.f, lj = 0.f;

    for (unsigned u = blockIdx.x * blockDim.x + threadIdx.x; u < NU; u += stride) {
        const unsigned b    = u >> 7;            // unit / (Z/4)
        const unsigned z4   = (u & 127u) << 2;   // starting z of this quad
        const unsigned base = b << 10;           // b * 2 * Z  (floats)

        // nat_params row b: interleaved {e0,e1} pairs -> two b128 NT loads
        const v4f n0 = ntload4(NP + base + (z4 << 1));
        const v4f n1 = ntload4(NP + base + (z4 << 1) + 4u);
        // T row b: T[b,0,z..] and T[b,1,z..] -> one b128 NT load each
        const v4f t0 = ntload4(T + base + z4);
        const v4f t1 = ntload4(T + base + ZDIM + z4);

        const float e0[4] = {n0.x, n0.z, n1.x, n1.z};
        const float e1[4] = {n0.y, n0.w, n1.y, n1.w};
        const float a0[4] = {t0.x, t0.y, t0.z, t0.w};
        const float a1[4] = {t1.x, t1.y, t1.z, t1.w};

        #pragma unroll
        for (int i = 0; i < 4; ++i) {
            tr = __builtin_fmaf(e0[i], a0[i], tr);
            tr = __builtin_fmaf(e1[i], a1[i], tr);
            lg += __builtin_amdgcn_logf(__builtin_fabsf(e0[i]));        // v_log_f32 (log2)
            const float q = 0.25f * e1[i] * e1[i];
            rt = __builtin_fmaf(q, __builtin_amdgcn_rcpf(e0[i]), rt);   // v_rcp_f32
        }
    }

    // log_jacobians reduction (256 KB, folded into the same pass)
    for (unsigned i = blockIdx.x * blockDim.x + threadIdx.x; i < BATCH; i += stride)
        lj += LJ[i];

    block_sum4(tr, lg, rt, lj, part + (size_t)blockIdx.x * 4u);
}

// Stage 2: one block reduces NBLK x 4 partials in fp64 and emits the 3 outputs.
__global__ void __launch_bounds__(TPB)
fe_stage2(const float* __restrict__ part, float* __restrict__ out) {
    __shared__ double sm[TPB][4];
    double a0 = 0.0, a1 = 0.0, a2 = 0.0, a3 = 0.0;
    for (unsigned r = threadIdx.x; r < NBLK; r += TPB) {
        const float* p = part + (size_t)r * 4u;
        a0 += (double)p[0]; a1 += (double)p[1]; a2 += (double)p[2]; a3 += (double)p[3];
    }
    sm[threadIdx.x][0] = a0; sm[threadIdx.x][1] = a1;
    sm[threadIdx.x][2] = a2; sm[threadIdx.x][3] = a3;
    __syncthreads();
    if (threadIdx.x == 0) {
        double t0 = 0.0, t1 = 0.0, t2 = 0.0, t3 = 0.0;
        for (int i = 0; i < TPB; ++i) {
            t0 += sm[i][0]; t1 += sm[i][1]; t2 += sm[i][2]; t3 += sm[i][3];
        }
        const double invB = 1.0 / (double)BATCH;
        const double LN2  = 0.6931471805599453;
        // t0 = trace, t1 = sum log2|eta0|, t2 = sum eta1^2/(4 eta0), t3 = sum lj
        out[0] = (float)((-0.5 * (t1 * LN2) - t2) * invB);  // log_normalizer
        out[1] = (float)(-t0 * invB);                        // neg_trace
        out[2] = (float)(-t3 * invB);                        // neg_log_det
    }
}

extern "C" void kernel_launch(void* const* d_in, const int* in_sizes, int n_in,
                              void* d_out, int out_size, void* d_ws, size_t ws_size,
                              hipStream_t stream) {
    const float* T  = (const float*)d_in[0];   // (B, 2, Z) f32
    const float* NP = (const float*)d_in[1];   // (B, Z, 2) f32
    const float* LJ = (const float*)d_in[2];   // (B,) f32
    float* part = (float*)d_ws;                // NBLK * 4 floats, fully rewritten each call
    float* out  = (float*)d_out;               // 3 floats

    fe_stage1<<<NBLK, TPB, 0, stream>>>(T, NP, LJ, part);
    fe_stage2<<<1, TPB, 0, stream>>>(part, out);
}